// SpecWaveTransformer_21114059227751
// MI455X (gfx1250) — compile-verified
//
#include <hip/hip_runtime.h>
#include <hip/hip_bf16.h>

// B=4, C=8, AM=4, NH=8, F=256, W=512, HD=32
// NP = B*AM*F*W = 2097152 floats per projection tensor

typedef __attribute__((ext_vector_type(2))) float v2f;
typedef __attribute__((ext_vector_type(8))) float v8f;

__device__ __forceinline__ v8f wmma_f32(v2f a, v2f b, v8f c) {
  // D = A(16x4 f32) * B(4x16 f32) + C(16x16 f32)
  return __builtin_amdgcn_wmma_f32_16x16x4_f32(
      /*neg_a=*/false, a, /*neg_b=*/false, b,
      /*c_mod=*/(short)0, c, /*reuse_a=*/false, /*reuse_b=*/false);
}

// ---------------------------------------------------------------------------
// Kernel 1: fused q/k/v 3x3 conv (SAME), C=8 -> AM=4. One thread per (b,f,w),
// computes all 12 outputs (3 proj x 4 channels). Weights staged in LDS.
// ---------------------------------------------------------------------------
__global__ __launch_bounds__(256) void conv3_kernel(
    const float* __restrict__ x,
    const float* __restrict__ Wq, const float* __restrict__ bq,
    const float* __restrict__ Wk, const float* __restrict__ bk,
    const float* __restrict__ Wv, const float* __restrict__ bv,
    float* __restrict__ yq, float* __restrict__ yk, float* __restrict__ yv) {
  __shared__ float wl[3 * 288];  // (AM=4)*(C=8)*9 = 288 per projection
  for (int i = threadIdx.x; i < 864; i += 256) {
    float v;
    if (i < 288)      v = Wq[i];
    else if (i < 576) v = Wk[i - 288];
    else              v = Wv[i - 576];
    wl[i] = v;
  }
  __syncthreads();

  int idx = blockIdx.x * 256 + threadIdx.x;  // over B*F*W = 524288
  int w = idx & 511;
  int f = (idx >> 9) & 255;
  int b = idx >> 17;

  float acc[12];
#pragma unroll
  for (int i = 0; i < 12; ++i) acc[i] = 0.f;

  for (int c = 0; c < 8; ++c) {
    const float* xc = x + ((size_t)(b * 8 + c)) * 131072;
#pragma unroll
    for (int df = 0; df < 3; ++df) {
      int ff = f + df - 1;
      if (ff < 0 || ff > 255) continue;
#pragma unroll
      for (int dw = 0; dw < 3; ++dw) {
        int ww = w + dw - 1;
        if (ww < 0 || ww > 511) continue;
        float xv = xc[(size_t)ff * 512 + ww];
        int wi = c * 9 + df * 3 + dw;
#pragma unroll
        for (int a = 0; a < 4; ++a) {
          acc[a]     += xv * wl[a * 72 + wi];
          acc[4 + a] += xv * wl[288 + a * 72 + wi];
          acc[8 + a] += xv * wl[576 + a * 72 + wi];
        }
      }
    }
  }
  size_t obase = ((size_t)b * 4) * 131072 + (size_t)f * 512 + w;
#pragma unroll
  for (int a = 0; a < 4; ++a) {
    yq[obase + (size_t)a * 131072] = acc[a] + bq[a];
    yk[obase + (size_t)a * 131072] = acc[4 + a] + bk[a];
    yv[obase + (size_t)a * 131072] = acc[8 + a] + bv[a];
  }
}

// ---------------------------------------------------------------------------
// Kernel 2: generic D[g,w] = sum_f A[inst%amod][g,f] * B[inst][f,w]
// M=256, K=256, N=512 per instance. Block = 4 waves, block tile 64x64,
// wave tile 16x64 (4 accumulators). Optional fused interleaved RoPE epilogue
// (dim = g&31, pos = w).
// ---------------------------------------------------------------------------
__global__ __launch_bounds__(128) void gemm256_kernel(
    const float* __restrict__ Amats,  // [amod][256][256]
    const float* __restrict__ Bmat,   // [nInst][256][512]
    float* __restrict__ Dmat,         // [nInst][256][512]
    int amod, int do_rope) {
  const int tid = threadIdx.x;
  const int wv = tid >> 5, lane = tid & 31;
  const int lm = lane & 15, half = lane >> 4;
  const int inst = blockIdx.z;
  const int g0 = blockIdx.y * 64 + wv * 16;
  const int w0 = blockIdx.x * 64;

  const float* A = Amats + (size_t)(inst % amod) * 65536;
  const float* Bp = Bmat + (size_t)inst * 131072;
  float* Dp = Dmat + (size_t)inst * 131072;

  v8f C[4] = {};
#pragma unroll 4
  for (int k0 = 0; k0 < 256; k0 += 4) {
    // A fragment: lane holds (M=g0+lm, K=k0+2*half .. +1) -> contiguous pair
    v2f a = *(const v2f*)(A + (size_t)(g0 + lm) * 256 + k0 + 2 * half);
    const float* bb = Bp + (size_t)(k0 + 2 * half) * 512;
#pragma unroll
    for (int n = 0; n < 4; ++n) {
      int w = w0 + 16 * n + lm;
      v2f b = {bb[w], bb[512 + w]};
      C[n] = wmma_f32(a, b, C[n]);
    }
  }

  const int rowbase = g0 + 8 * half;  // M = rowbase + e  (e = accumulator elem)
  if (do_rope) {
#pragma unroll
    for (int n = 0; n < 4; ++n) {
      float pos = (float)(w0 + 16 * n + lm);
#pragma unroll
      for (int j = 0; j < 4; ++j) {
        int hd = (rowbase + 2 * j) & 31;               // even dim index 2i
        float inv = __powf(10000.f, -(float)hd / 32.f);  // 10000^{-2i/d}
        float s, cs;
        __sincosf(pos * inv, &s, &cs);
        float x1 = C[n][2 * j], x2 = C[n][2 * j + 1];
        C[n][2 * j]     = x1 * cs - x2 * s;
        C[n][2 * j + 1] = x2 * cs + x1 * s;
      }
    }
  }
#pragma unroll
  for (int n = 0; n < 4; ++n) {
    int w = w0 + 16 * n + lm;
#pragma unroll
    for (int e = 0; e < 8; ++e)
      Dp[(size_t)(rowbase + e) * 512 + w] = C[n][e];
  }
}

// ---------------------------------------------------------------------------
// Kernel 3: qk[head, qpos, kpos] = (1/16) * sum_hd q[hd,qpos] * k[hd,kpos]
// head = inst*8 + nh. Per-head GEMM M=N=512, K=32. Writes logits to d_out.
// ---------------------------------------------------------------------------
__global__ __launch_bounds__(128) void qk_kernel(
    const float* __restrict__ qst, const float* __restrict__ kst,
    float* __restrict__ qk) {
  const int tid = threadIdx.x;
  const int wv = tid >> 5, lane = tid & 31;
  const int lm = lane & 15, half = lane >> 4;
  const int head = blockIdx.z;  // 0..127
  const int inst = head >> 3, nh = head & 7;
  const int q0 = blockIdx.y * 64 + wv * 16;
  const int kc0 = blockIdx.x * 64;

  const float* Qp = qst + (size_t)inst * 131072 + (size_t)nh * 32 * 512;
  const float* Kp = kst + (size_t)inst * 131072 + (size_t)nh * 32 * 512;

  v8f C[4] = {};
#pragma unroll
  for (int k0 = 0; k0 < 32; k0 += 4) {
    const float* qrow = Qp + (size_t)(k0 + 2 * half) * 512;
    const float* krow = Kp + (size_t)(k0 + 2 * half) * 512;
    v2f a = {qrow[q0 + lm], qrow[512 + q0 + lm]};
#pragma unroll
    for (int n = 0; n < 4; ++n) {
      int kc = kc0 + 16 * n + lm;
      v2f b = {krow[kc], krow[512 + kc]};
      C[n] = wmma_f32(a, b, C[n]);
    }
  }

  float* out = qk + (size_t)head * 512 * 512;
  const int rowbase = q0 + 8 * half;
#pragma unroll
  for (int n = 0; n < 4; ++n) {
    int kc = kc0 + 16 * n + lm;
#pragma unroll
    for (int e = 0; e < 8; ++e)
      out[(size_t)(rowbase + e) * 512 + kc] = C[n][e] * 0.0625f;  // 1/sqrt(256)
  }
}

// ---------------------------------------------------------------------------
// Kernel 4: per-row softmax stats (max, sumexp). One wave32 per row.
// ---------------------------------------------------------------------------
__global__ __launch_bounds__(128) void softmax_stats_kernel(
    const float* __restrict__ qk, float* __restrict__ stats) {
  int row = blockIdx.x * 4 + (threadIdx.x >> 5);  // 128*512 = 65536 rows
  int lane = threadIdx.x & 31;
  const float* r = qk + (size_t)row * 512;
  float m = -3.4e38f;
  for (int j = lane; j < 512; j += 32) m = fmaxf(m, r[j]);
#pragma unroll
  for (int off = 16; off > 0; off >>= 1) m = fmaxf(m, __shfl_xor(m, off, 32));
  float s = 0.f;
  for (int j = lane; j < 512; j += 32) s += __expf(r[j] - m);
#pragma unroll
  for (int off = 16; off > 0; off >>= 1) s += __shfl_xor(s, off, 32);
  if (lane == 0) {
    stats[2 * row] = m;
    stats[2 * row + 1] = s;
  }
}

// ---------------------------------------------------------------------------
// Kernel 5: a[qpos,hd] = sum_k softmax(qk)[qpos,k] * v[k,hd].
// Probabilities rebuilt on-the-fly in the A-fragment load (exp + 1/sum fused).
// Wave tile: 16 qpos x 32 hd (2 accumulators). Output to astore[g=nh*32+hd][qpos].
// ---------------------------------------------------------------------------
__global__ __launch_bounds__(128) void av_kernel(
    const float* __restrict__ qk, const float* __restrict__ stats,
    const float* __restrict__ vst, float* __restrict__ ast) {
  const int tid = threadIdx.x;
  const int wv = tid >> 5, lane = tid & 31;
  const int lm = lane & 15, half = lane >> 4;
  const int head = blockIdx.y;
  const int inst = head >> 3, nh = head & 7;
  const int q0 = blockIdx.x * 64 + wv * 16;

  const int row = head * 512 + q0 + lm;
  const float* qkrow = qk + (size_t)row * 512;
  const float rmax = stats[2 * row];
  const float rinv = 1.f / stats[2 * row + 1];
  const float* Vp = vst + (size_t)inst * 131072 + (size_t)nh * 32 * 512;

  v8f C[2] = {};
  for (int k0 = 0; k0 < 512; k0 += 4) {
    int kk = k0 + 2 * half;
    v2f a = {__expf(qkrow[kk] - rmax) * rinv,
             __expf(qkrow[kk + 1] - rmax) * rinv};
#pragma unroll
    for (int n = 0; n < 2; ++n) {
      const float* vrow = Vp + (size_t)(n * 16 + lm) * 512 + kk;  // v[kpos,hd] = vst[g][kpos]
      v2f b = {vrow[0], vrow[1]};
      C[n] = wmma_f32(a, b, C[n]);
    }
  }
#pragma unroll
  for (int n = 0; n < 2; ++n) {
    int g = nh * 32 + n * 16 + lm;
    float* dst = ast + (size_t)inst * 131072 + (size_t)g * 512 + q0 + 8 * half;
#pragma unroll
    for (int e = 0; e < 8; ++e) dst[e] = C[n][e];  // consecutive qpos -> contiguous
  }
}

// ---------------------------------------------------------------------------
// Kernel 6: depthwise mix: out1[b,o,f,w] = sum_{c<12} concat(x,a)[b,c,f,w]*Wdw[o,c]
// ---------------------------------------------------------------------------
__global__ __launch_bounds__(256) void dw_kernel(
    const float* __restrict__ x, const float* __restrict__ ast,
    const float* __restrict__ Wdw, float* __restrict__ out1) {
  int idx = blockIdx.x * 256 + threadIdx.x;  // over B*F*W
  int w = idx & 511;
  int f = (idx >> 9) & 255;
  int b = idx >> 17;
  float xa[12];
#pragma unroll
  for (int c = 0; c < 8; ++c)
    xa[c] = x[(((size_t)b * 8 + c) * 256 + f) * 512 + w];
#pragma unroll
  for (int a = 0; a < 4; ++a)
    xa[8 + a] = ast[(((size_t)b * 4 + a) * 256 + f) * 512 + w];
#pragma unroll
  for (int o = 0; o < 8; ++o) {
    float acc = 0.f;
#pragma unroll
    for (int c = 0; c < 12; ++c) acc += xa[c] * Wdw[o * 12 + c];
    out1[(((size_t)b * 8 + o) * 256 + f) * 512 + w] = acc;
  }
}

// ---------------------------------------------------------------------------
extern "C" void kernel_launch(void* const* d_in, const int* in_sizes, int n_in,
                              void* d_out, int out_size, void* d_ws,
                              size_t ws_size, hipStream_t stream) {
  const float* x       = (const float*)d_in[0];
  const float* Wq_conv = (const float*)d_in[1];
  const float* bq_conv = (const float*)d_in[2];
  const float* Wq_ml   = (const float*)d_in[3];
  const float* Wk_conv = (const float*)d_in[4];
  const float* bk_conv = (const float*)d_in[5];
  const float* Wk_ml   = (const float*)d_in[6];
  const float* Wv_conv = (const float*)d_in[7];
  const float* bv_conv = (const float*)d_in[8];
  const float* Wv_ml   = (const float*)d_in[9];
  const float* Wo_dw   = (const float*)d_in[10];
  const float* Wo_pw   = (const float*)d_in[11];

  float* out = (float*)d_out;        // (4,8,256,512) = 4194304 floats
  float* qk  = out + 4194304;        // (4,4,8,512,512) = 33554432 floats

  const size_t NP = 2097152;         // B*AM*F*W
  float* ws    = (float*)d_ws;
  float* convQ = ws;                 // [0, NP)
  float* convK = ws + NP;            // [NP, 2NP)
  float* convV = ws + 2 * NP;        // [2NP, 3NP)
  float* qst   = ws + 3 * NP;
  float* kst   = ws + 4 * NP;
  float* vst   = ws + 5 * NP;
  float* stats = ws + 6 * NP;        // 131072 floats
  float* astore = convQ;             // conv buffers dead after projections
  float* out1   = convK;             // needs 2*NP: [NP, 3NP)

  // 1) fused convs
  conv3_kernel<<<2048, 256, 0, stream>>>(x, Wq_conv, bq_conv, Wk_conv, bk_conv,
                                         Wv_conv, bv_conv, convQ, convK, convV);
  // 2) projection GEMMs (RoPE fused for q, k)
  dim3 gp(8, 4, 16);
  gemm256_kernel<<<gp, 128, 0, stream>>>(Wq_ml, convQ, qst, 4, 1);
  gemm256_kernel<<<gp, 128, 0, stream>>>(Wk_ml, convK, kst, 4, 1);
  gemm256_kernel<<<gp, 128, 0, stream>>>(Wv_ml, convV, vst, 4, 0);
  // 3) attention logits -> d_out (second output)
  qk_kernel<<<dim3(8, 8, 128), 128, 0, stream>>>(qst, kst, qk);
  // 4) softmax row stats
  softmax_stats_kernel<<<16384, 128, 0, stream>>>(qk, stats);
  // 5) P*V with fused exp-normalize
  av_kernel<<<dim3(8, 128), 128, 0, stream>>>(qk, stats, vst, astore);
  // 6) depthwise channel mix over concat(x, a)
  dw_kernel<<<2048, 256, 0, stream>>>(x, astore, Wo_dw, out1);
  // 7) output pointwise GEMM -> d_out (first output)
  gemm256_kernel<<<dim3(8, 4, 32), 128, 0, stream>>>(Wo_pw, out1, out, 8, 0);
}